// BiLSTM_CRF_13391708029875
// MI455X (gfx1250) — compile-verified
//
#include <hip/hip_runtime.h>
#include <math.h>
#include <stdint.h>

// Problem constants (from reference)
#define TT     2048   // sentence length
#define LW     20     // max word length
#define CED    25     // char embed dim
#define COD    25     // char cnn out channels
#define EW     300    // word embed dim
#define DIN    325    // E + CO (LSTM input dim)
#define DPAD   352    // DIN padded to multiple of 32 (and 4) for WMMA K loop
#define HH     256    // hidden
#define G4     1024   // 4*H
#define KK     12     // tags
#define NEGV   (-10000.0f)
#define STARTT 10
#define STOPT  11

// LSTM W_hh LDS cache: first NROWS_LDS gate rows resident in LDS, padded rows
// (260-float stride => conflict-free ds_load_b128 across a wave: 260 % 64 == 4,
// lanes 0-15 of a b128 phase touch all 64 banks exactly once).
#define NROWS_LDS  288
#define WPAD       260                      // floats per cached row
#define LSTM_SMEM_FLOATS (G4 + HH + NROWS_LDS * WPAD)
#define LSTM_SMEM_BYTES  (LSTM_SMEM_FLOATS * 4)

typedef __attribute__((ext_vector_type(2))) float v2f;
typedef __attribute__((ext_vector_type(8))) float v8f;
typedef __attribute__((ext_vector_type(4))) unsigned int v4u;
typedef __attribute__((ext_vector_type(4))) int v4i;
typedef __attribute__((ext_vector_type(8))) int v8i;

#if defined(__gfx1250__) && __has_builtin(__builtin_amdgcn_tensor_load_to_lds) && \
    __has_builtin(__builtin_amdgcn_s_wait_tensorcnt)
#define USE_TDM 1
#else
#define USE_TDM 0
#endif

__device__ __forceinline__ float sigmoidf_(float x) { return 1.0f / (1.0f + expf(-x)); }

// ---------------------------------------------------------------------------
// Kernel 1: char CNN (+maxpool) and word-emb gather -> x [TT, DPAD] (zero-padded)
// ---------------------------------------------------------------------------
__global__ void build_features(const int* __restrict__ sentence,
                               const int* __restrict__ chars,
                               const float* __restrict__ word_emb,
                               const float* __restrict__ char_emb,
                               const float* __restrict__ conv_w,  // [CO,1,3,CE]
                               const float* __restrict__ conv_b,  // [CO]
                               float* __restrict__ x) {           // [TT, DPAD]
    __shared__ float ce[(LW + 4) * CED];  // 2 zero rows of padding each side
    const int t   = blockIdx.x;
    const int tid = threadIdx.x;

    for (int i = tid; i < (LW + 4) * CED; i += blockDim.x) ce[i] = 0.0f;
    __syncthreads();
    for (int i = tid; i < LW * CED; i += blockDim.x) {
        const int ci = i / CED, c = i % CED;
        ce[(ci + 2) * CED + c] = char_emb[chars[t * LW + ci] * CED + c];
    }
    __syncthreads();

    if (tid < COD) {
        const float* w = conv_w + tid * 3 * CED;
        const float  b = conv_b[tid];
        float best = -3.0e38f;
        for (int p = 0; p < LW + 2; ++p) {
            float s = b;
#pragma unroll
            for (int kh = 0; kh < 3; ++kh)
                for (int c = 0; c < CED; ++c)
                    s += ce[(p + kh) * CED + c] * w[kh * CED + c];
            best = fmaxf(best, s);
        }
        x[t * DPAD + EW + tid] = best;  // concat: [word(300) | char(25)]
    }

    const int s = sentence[t];
    for (int e = tid; e < EW; e += blockDim.x)
        x[t * DPAD + e] = word_emb[s * EW + e];
    for (int e = EW + COD + tid; e < DPAD; e += blockDim.x)
        x[t * DPAD + e] = 0.0f;  // zero pad tail so WMMA K loop is clean
}

// ---------------------------------------------------------------------------
// Kernel 2: input-projection GEMM with FP32 WMMA (v_wmma_f32_16x16x4_f32):
//   G_dir = x[TT,DPAD] @ W_ih_dir^T + b_dir   -> [TT, G4]
// One wave computes one 16x16 tile; block = 4 waves; grid.z = direction.
// ---------------------------------------------------------------------------
__global__ void input_gemm(const float* __restrict__ x,
                           const float* __restrict__ w_ih_f, const float* __restrict__ b_f,
                           const float* __restrict__ w_ih_b, const float* __restrict__ b_b,
                           float* __restrict__ G_f, float* __restrict__ G_b) {
    const float* W = blockIdx.z ? w_ih_b : w_ih_f;  // [G4, DIN] row-major
    const float* B = blockIdx.z ? b_b    : b_f;
    float*       G = blockIdx.z ? G_b    : G_f;

    const int lane = threadIdx.x & 31;
    const int wave = threadIdx.x >> 5;
    const int tm   = blockIdx.x * 16;
    const int tn   = (blockIdx.y * 4 + wave) * 16;

    const int    mn   = lane & 15;           // row of A / col of B handled by lane
    const int    koff = (lane >> 4) << 1;    // 0 for lanes 0-15, 2 for lanes 16-31
    const float* xrow = x + (size_t)(tm + mn) * DPAD;
    const float* wrow = W + (size_t)(tn + mn) * DIN;

    v8f acc = {};
    int k0 = 0;
    for (; k0 < 320; k0 += 4) {              // k < 320: no bounds checks
        v2f a, b;
        a.x = xrow[k0 + koff];
        a.y = xrow[k0 + koff + 1];
        b.x = wrow[k0 + koff];
        b.y = wrow[k0 + koff + 1];
        acc = __builtin_amdgcn_wmma_f32_16x16x4_f32(false, a, false, b,
                                                    (short)0, acc, false, false);
    }
    for (; k0 < DPAD; k0 += 4) {             // tail: guard W reads past DIN
        v2f a, b;
        a.x = xrow[k0 + koff];
        a.y = xrow[k0 + koff + 1];
        b.x = (k0 + koff     < DIN) ? wrow[k0 + koff]     : 0.0f;
        b.y = (k0 + koff + 1 < DIN) ? wrow[k0 + koff + 1] : 0.0f;
        acc = __builtin_amdgcn_wmma_f32_16x16x4_f32(false, a, false, b,
                                                    (short)0, acc, false, false);
    }

    const int   rbase = (lane >> 4) * 8;     // C/D: VGPR r -> M = r + 8*(lane>=16)
    const float bias  = B[tn + mn];
#pragma unroll
    for (int r = 0; r < 8; ++r)
        G[(size_t)(tm + rbase + r) * G4 + tn + mn] = acc[r] + bias;
}

// ---------------------------------------------------------------------------
// TDM: DMA the first `rows` rows of W_hh [G4, HH] into LDS at byte offset
// lds_off, inserting 4 DWORDs of padding every 256 DWORDs (=> 260-float rows).
// Issued by one wave; completion via s_wait_tensorcnt.
// ---------------------------------------------------------------------------
#if USE_TDM
__device__ __forceinline__ void tdm_load_whh(const float* Whh, uint32_t lds_off,
                                             uint32_t rows) {
    const uint64_t ga = (uint64_t)(uintptr_t)Whh;
    v4u g0;
    g0.x = 1u;                                            // count=1, user descriptor
    g0.y = lds_off;                                       // lds_addr (bytes)
    g0.z = (uint32_t)(ga & 0xffffffffu);                  // global_addr[31:0]
    g0.w = (uint32_t)((ga >> 32) & 0x01ffffffu) | (2u << 30);  // addr[56:32] | type=2

    const uint32_t td0 = HH, td1 = G4;                    // tensor dims (elements)
    const uint32_t tile0 = HH, tile1 = rows, tile2 = 0;   // tile dims
    const uint64_t s0 = HH;                               // tensor_dim0_stride
    const uint64_t s1 = 0;

    v8i g1;
    // w0: data_size=2 (4B) | pad_enable | pad_interval=7 (256 dw) | pad_amount=3 (4 dw)
    g1[0] = (int)((2u << 16) | (1u << 20) | (7u << 22) | (3u << 25));
    g1[1] = (int)((td0 & 0xffffu) << 16);
    g1[2] = (int)(((td0 >> 16) & 0xffffu) | ((td1 & 0xffffu) << 16));
    g1[3] = (int)(((td1 >> 16) & 0xffffu) | (tile0 << 16));
    g1[4] = (int)(tile1 | (tile2 << 16));
    g1[5] = (int)(s0 & 0xffffffffu);
    g1[6] = (int)(((s0 >> 32) & 0xffffu) | ((uint32_t)(s1 & 0xffffu) << 16));
    g1[7] = (int)((s1 >> 16) & 0xffffffffu);

    const v4i z4 = {0, 0, 0, 0};
#if __clang_major__ >= 23
    const v8i z8 = {0, 0, 0, 0, 0, 0, 0, 0};
    __builtin_amdgcn_tensor_load_to_lds(g0, g1, z4, z4, z8, 0);
#else
    __builtin_amdgcn_tensor_load_to_lds(g0, g1, z4, z4, 0);
#endif
    __builtin_amdgcn_s_wait_tensorcnt((short)0);
}
#endif

// ---------------------------------------------------------------------------
// Kernel 3: persistent LSTM recurrence. One block per direction, 1024 threads.
// Thread r owns gate row r: acc = G[t,r] + dot(W_hh[r,:], h).
// First NROWS_LDS rows of W_hh are TDM-staged into LDS; rest stream from L2.
// ---------------------------------------------------------------------------
__global__ void __launch_bounds__(1024) lstm_persistent(
        const float* __restrict__ G_f, const float* __restrict__ G_b,
        const float* __restrict__ w_hh_f, const float* __restrict__ w_hh_b,
        float* __restrict__ hs_f, float* __restrict__ hs_b) {
    extern __shared__ float smem[];
    float* gates  = smem;            // [G4]
    float* h_lds  = smem + G4;       // [HH]
    float* wcache = smem + G4 + HH;  // [NROWS_LDS * WPAD]

    const int dir = blockIdx.x;
    const float* G   = dir ? G_b    : G_f;
    const float* Whh = dir ? w_hh_b : w_hh_f;   // [G4, HH] row-major
    float*       hs  = dir ? hs_b   : hs_f;     // [TT, HH]

    const int r = threadIdx.x;

#if USE_TDM
    if (r < 32) {  // wave 0 issues the DMA and waits for it
        tdm_load_whh(Whh, (uint32_t)(uintptr_t)wcache, NROWS_LDS);
    }
#else
    // fallback: cooperative staged copy with the same padded layout
    for (int i = r; i < NROWS_LDS * HH; i += 1024) {
        const int row = i / HH, col = i % HH;
        wcache[row * WPAD + col] = Whh[row * HH + col];
    }
#endif
    if (r < HH) h_lds[r] = 0.0f;
    float c = 0.0f;
    const float* wr = Whh + (size_t)r * HH;
    if (r >= NROWS_LDS) __builtin_prefetch(wr, 0, 0);   // global_prefetch_b8
    __syncthreads();

    for (int step = 0; step < TT; ++step) {
        const int t = dir ? (TT - 1 - step) : step;
        float acc = G[(size_t)t * G4 + r];      // includes W_ih@x + b
        const float4* h4 = (const float4*)h_lds;
        if (r < NROWS_LDS) {
            const float4* w4 = (const float4*)(wcache + (size_t)r * WPAD);
#pragma unroll 8
            for (int k = 0; k < HH / 4; ++k) {
                const float4 w = w4[k];
                const float4 h = h4[k];
                acc += w.x * h.x + w.y * h.y + w.z * h.z + w.w * h.w;
            }
        } else {
            const float4* w4 = (const float4*)wr;
#pragma unroll 8
            for (int k = 0; k < HH / 4; ++k) {
                const float4 w = w4[k];
                const float4 h = h4[k];
                acc += w.x * h.x + w.y * h.y + w.z * h.z + w.w * h.w;
            }
        }
        gates[r] = acc;
        __syncthreads();                        // all gates visible; h reads done
        if (r < HH) {
            const float ig = sigmoidf_(gates[r]);
            const float fg = sigmoidf_(gates[r + HH]);
            const float gg = tanhf(gates[r + 2 * HH]);
            const float og = sigmoidf_(gates[r + 3 * HH]);
            c = fg * c + ig * gg;
            const float h = og * tanhf(c);
            hs[(size_t)t * HH + r] = h;
            h_lds[r] = h;
        }
        __syncthreads();                        // publish new h before next step
    }
}

// ---------------------------------------------------------------------------
// Kernel 4: feats[t,k] = b[k] + [hf(t)|hb(t)] . h2t_w[k,:]
// ---------------------------------------------------------------------------
__global__ void h2t_proj(const float* __restrict__ hs_f, const float* __restrict__ hs_b,
                         const float* __restrict__ W,     // [KK, 2*HH]
                         const float* __restrict__ b,     // [KK]
                         float* __restrict__ feats) {     // [TT, KK]
    const int gid = blockIdx.x * blockDim.x + threadIdx.x;
    if (gid >= TT * KK) return;
    const int t = gid / KK, k = gid % KK;
    const float* wf = W + (size_t)k * (2 * HH);
    const float* wb = wf + HH;
    const float* hf = hs_f + (size_t)t * HH;
    const float* hb = hs_b + (size_t)t * HH;
    float s = b[k];
    for (int j = 0; j < HH; ++j) s += hf[j] * wf[j] + hb[j] * wb[j];
    feats[gid] = s;
}

// ---------------------------------------------------------------------------
// Kernel 5: Viterbi decode + backtrack. Single wave, lane i = tag i.
// ---------------------------------------------------------------------------
__global__ void viterbi(const float* __restrict__ feats,  // [TT, KK]
                        const float* __restrict__ trans,  // [KK, KK] (to x from)
                        int* __restrict__ bp,             // [TT, KK] scratch
                        float* __restrict__ out) {        // [1 + TT]
    __shared__ float fv[KK];
    __shared__ float tr[KK * KK];
    const int tid = threadIdx.x;
    for (int i = tid; i < KK * KK; i += blockDim.x) tr[i] = trans[i];
    if (tid < KK) fv[tid] = (tid == STARTT) ? 0.0f : NEGV;
    __syncthreads();

    for (int t = 0; t < TT; ++t) {
        float nv = 0.0f;
        if (tid < KK) {
            float best = -3.4e38f; int bj = 0;
            for (int j = 0; j < KK; ++j) {
                const float v = fv[j] + tr[tid * KK + j];
                if (v > best) { best = v; bj = j; }
            }
            bp[t * KK + tid] = bj;
            nv = best + feats[t * KK + tid];
        }
        __syncthreads();
        if (tid < KK) fv[tid] = nv;
        __syncthreads();
    }

    if (tid == 0) {
        float best = -3.4e38f; int bi = 0;
        for (int i = 0; i < KK; ++i) {
            float v = fv[i] + tr[STOPT * KK + i];
            if (i == STARTT || i == STOPT) v = NEGV;
            if (v > best) { best = v; bi = i; }
        }
        out[0] = best;
        int tag = bi;
        for (int t = TT - 1; t >= 0; --t) {
            out[1 + t] = (float)tag;
            tag = bp[t * KK + tag];
        }
    }
}

// ---------------------------------------------------------------------------
extern "C" void kernel_launch(void* const* d_in, const int* in_sizes, int n_in,
                              void* d_out, int out_size, void* d_ws, size_t ws_size,
                              hipStream_t stream) {
    const int*   sentence = (const int*)  d_in[0];
    const int*   chars    = (const int*)  d_in[1];
    const float* word_emb = (const float*)d_in[2];
    const float* char_emb = (const float*)d_in[3];
    const float* conv_w   = (const float*)d_in[4];
    const float* conv_b   = (const float*)d_in[5];
    const float* w_ih_f   = (const float*)d_in[6];
    const float* w_hh_f   = (const float*)d_in[7];
    const float* b_f      = (const float*)d_in[8];
    const float* w_ih_b   = (const float*)d_in[9];
    const float* w_hh_b   = (const float*)d_in[10];
    const float* b_b      = (const float*)d_in[11];
    const float* h2t_w    = (const float*)d_in[12];
    const float* h2t_b    = (const float*)d_in[13];
    const float* trans    = (const float*)d_in[14];
    float* out = (float*)d_out;

    char*  ws  = (char*)d_ws;
    size_t off = 0;
    auto carve = [&](size_t bytes) -> void* {
        void* p = ws + off;
        off += (bytes + 255) & ~(size_t)255;
        return p;
    };
    float* x     = (float*)carve((size_t)TT * DPAD * sizeof(float));
    float* G_f   = (float*)carve((size_t)TT * G4 * sizeof(float));
    float* G_b   = (float*)carve((size_t)TT * G4 * sizeof(float));
    float* hs_f  = (float*)carve((size_t)TT * HH * sizeof(float));
    float* hs_b  = (float*)carve((size_t)TT * HH * sizeof(float));
    float* feats = (float*)carve((size_t)TT * KK * sizeof(float));
    int*   bp    = (int*)  carve((size_t)TT * KK * sizeof(int));

    build_features<<<TT, 128, 0, stream>>>(sentence, chars, word_emb, char_emb,
                                           conv_w, conv_b, x);
    input_gemm<<<dim3(TT / 16, G4 / 64, 2), 128, 0, stream>>>(
        x, w_ih_f, b_f, w_ih_b, b_b, G_f, G_b);
    lstm_persistent<<<2, 1024, LSTM_SMEM_BYTES, stream>>>(
        G_f, G_b, w_hh_f, w_hh_b, hs_f, hs_b);
    h2t_proj<<<(TT * KK + 255) / 256, 256, 0, stream>>>(hs_f, hs_b, h2t_w, h2t_b, feats);
    viterbi<<<1, 32, 0, stream>>>(feats, trans, bp, out);
}